// MoeGate_45148696217035
// MI455X (gfx1250) — compile-verified
//
#include <hip/hip_runtime.h>
#include <hip/hip_bf16.h>
#include <math.h>

// ---------------------------------------------------------------------------
// MoE gate for MI455X (gfx1250, wave32):
//   routing = x @ W + b + bias        (fp32 GEMM via V_WMMA_F32_16X16X4_F32)
//   x streamed HBM -> LDS with double-buffered ASYNC loads (ASYNCcnt pipelined)
//   top-8 per token + softmax         (LDS-resident stable insertion select)
//   expert histogram                  (ds_add_u32 -> global_atomic_add_f32)
//   new_total / new_bias update       (tiny finalize kernel)
// ---------------------------------------------------------------------------

typedef __attribute__((ext_vector_type(2))) float v2f;
typedef __attribute__((ext_vector_type(4))) float v4f;
typedef __attribute__((ext_vector_type(8))) float v8f;
typedef __attribute__((ext_vector_type(4))) int   v4i;

// async-load pointer types: AS1 = global, AS3 = LDS
typedef __attribute__((address_space(1))) v4i g_v4i;
typedef __attribute__((address_space(3))) v4i l_v4i;

#define N_EMBD   2048
#define N_EXP    64
#define TOPK     8
#define TPB      64          // tokens per block (4 M-tiles of 16)
#define KB       64          // K chunk staged in LDS
#define NCHUNK   (N_EMBD / KB)
#define LSTRIDE  68          // floats: 68 % 64 == 4 -> bank-conflict-free; 68 % 4 == 0 -> 16B aligned rows
#define BIAS_LR  0.1f

#if __has_builtin(__builtin_amdgcn_global_load_async_to_lds_b128) && \
    __has_builtin(__builtin_amdgcn_s_wait_asynccnt)
#define HAS_ASYNC_LDS 1
#else
#define HAS_ASYNC_LDS 0
#endif

__global__ __launch_bounds__(128)
void moe_gate_gemm_topk(const float* __restrict__ x,
                        const float* __restrict__ W,
                        const float* __restrict__ b,
                        const float* __restrict__ bias,
                        float* __restrict__ out_probs,
                        float* __restrict__ out_idx,
                        float* __restrict__ counts)
{
    __shared__ float lds_x[2][TPB * LSTRIDE];   // ping-pong x tiles; buf0 reused as routing tile
    __shared__ int   hist[N_EXP];

    const int tid  = threadIdx.x;
    const int wave = tid >> 5;             // 4 waves, wave32
    const int lane = tid & 31;
    const int lo   = lane & 15;            // N / M-row sub-index per WMMA layout
    const int hi   = lane >> 4;            // K / M-half selector per WMMA layout
    const int n0   = wave * 16;            // this wave's 16-expert N-tile
    const int m0   = blockIdx.x * TPB;     // first token of this block

    if (tid < N_EXP) hist[tid] = 0;

    // ---- stage one 64x64 f32 chunk of x into LDS buffer `buf` -------------
    auto issue_chunk = [&](int kc, int buf) {
        #pragma unroll
        for (int i = 0; i < 8; ++i) {
            int idx4 = tid + i * 128;                  // 1024 float4 total
            int row  = idx4 >> 4;                      // 16 float4 per row
            int col  = (idx4 & 15) << 2;
            const float* gp = x + (size_t)(m0 + row) * N_EMBD + kc + col;
            float*       lp = &lds_x[buf][row * LSTRIDE + col];
#if HAS_ASYNC_LDS
            // global_load_async_to_lds_b128: per-lane 16B HBM->LDS, no VGPR staging,
            // tracked by ASYNCcnt so it overlaps the WMMA loop on the other buffer.
            __builtin_amdgcn_global_load_async_to_lds_b128(
                (g_v4i*)gp, (l_v4i*)lp, /*offset=*/0, /*cpol=*/0);
#else
            *(v4f*)lp = *(const v4f*)gp;
#endif
        }
    };

    v8f acc[4];
    #pragma unroll
    for (int mt = 0; mt < 4; ++mt)
        #pragma unroll
        for (int r = 0; r < 8; ++r)
            acc[mt][r] = 0.0f;

    issue_chunk(0, 0);                                 // prologue: chunk 0 -> buf 0

    for (int kci = 0; kci < NCHUNK; ++kci) {
        const int cur = kci & 1;
        __syncthreads();                               // all waves done reading buf cur^1 (prev compute)
        if (kci + 1 < NCHUNK) {
            issue_chunk((kci + 1) * KB, cur ^ 1);      // prefetch next chunk into other buffer
#if HAS_ASYNC_LDS
            __builtin_amdgcn_s_wait_asynccnt(8);       // in-order: oldest 8 (chunk kci) retired
#endif
        } else {
#if HAS_ASYNC_LDS
            __builtin_amdgcn_s_wait_asynccnt(0);
#endif
        }
        __syncthreads();                               // everyone's chunk-kci LDS writes visible

        // ---- WMMA over this K chunk: D(16x16) = A(16x4) * B(4x16) + C ----
        // A frag (fp32 16x4): v0 = x[M=lo][k + 2*hi], v1 = x[M=lo][k + 2*hi + 1]
        // B frag (fp32 4x16): v0 = W[k + 2*hi][n0+lo], v1 = W[k + 2*hi + 1][n0+lo]
        const int kc = kci * KB;
        #pragma unroll 4
        for (int kk = 0; kk < KB; kk += 4) {
            const int kg = kc + kk + 2 * hi;
            v2f bf;
            bf.x = W[(size_t)kg * N_EXP + n0 + lo];
            bf.y = W[(size_t)(kg + 1) * N_EXP + n0 + lo];
            #pragma unroll
            for (int mt = 0; mt < 4; ++mt) {
                v2f af = *(const v2f*)(&lds_x[cur][(mt * 16 + lo) * LSTRIDE + kk + 2 * hi]);
                acc[mt] = __builtin_amdgcn_wmma_f32_16x16x4_f32(
                    /*neg_a=*/false, af, /*neg_b=*/false, bf,
                    /*c_mod=*/(short)0, acc[mt],
                    /*reuse_a=*/false, /*reuse_b=*/false);
            }
        }
    }

    // ---- write routing tile to LDS buf0 (+ linear bias terms) ----
    const float badd = b[n0 + lo] + bias[n0 + lo];
    __syncthreads();
    float* ldsr = &lds_x[0][0];
    #pragma unroll
    for (int mt = 0; mt < 4; ++mt)
        #pragma unroll
        for (int r = 0; r < 8; ++r)     // C layout: VGPR r, lane: row = r + 8*hi, col = lo
            ldsr[(mt * 16 + r + 8 * hi) * LSTRIDE + n0 + lo] = acc[mt][r] + badd;
    __syncthreads();

    // ---- per-token top-8 (stable: ties keep lower expert index) + softmax ----
    if (tid < TPB) {
        float v[TOPK]; int ix[TOPK];
        #pragma unroll
        for (int j = 0; j < TOPK; ++j) { v[j] = -INFINITY; ix[j] = 0; }
        const float* row = &ldsr[tid * LSTRIDE];
        for (int e = 0; e < N_EXP; ++e) {
            float r = row[e];
            if (r > v[TOPK - 1]) {
                v[TOPK - 1] = r; ix[TOPK - 1] = e;
                #pragma unroll
                for (int s = TOPK - 1; s >= 1; --s) {
                    if (v[s] > v[s - 1]) {
                        float tv = v[s]; v[s] = v[s - 1]; v[s - 1] = tv;
                        int   ti = ix[s]; ix[s] = ix[s - 1]; ix[s - 1] = ti;
                    }
                }
            }
        }
        float mx = v[0], p[TOPK], s = 0.0f;
        #pragma unroll
        for (int j = 0; j < TOPK; ++j) { p[j] = __expf(v[j] - mx); s += p[j]; }
        const float inv = 1.0f / s;
        const size_t gtok = (size_t)(m0 + tid);
        #pragma unroll
        for (int j = 0; j < TOPK; ++j) {
            out_probs[gtok * TOPK + j] = p[j] * inv;
            out_idx  [gtok * TOPK + j] = (float)ix[j];
            atomicAdd(&hist[ix[j]], 1);                 // ds_add_u32
        }
    }
    __syncthreads();
    if (tid < N_EXP) {
        int c = hist[tid];
        if (c) atomicAdd(counts + tid, (float)c);       // global_atomic_add_f32 (exact: int-valued)
    }
}

__global__ void moe_gate_zero(float* __restrict__ counts)
{
    counts[threadIdx.x] = 0.0f;
}

__global__ void moe_gate_finalize(const float* __restrict__ total_in,
                                  const float* __restrict__ bias,
                                  const float* __restrict__ counts,
                                  float* __restrict__ out_total,
                                  float* __restrict__ out_bias)
{
    __shared__ float s_nt[N_EXP];
    const int e = threadIdx.x;
    const float nt = total_in[e] + counts[e];
    s_nt[e] = nt;
    __syncthreads();
    float s = 0.0f;
    #pragma unroll
    for (int i = 0; i < N_EXP; ++i) s += s_nt[i];
    const float mean = s * (1.0f / (float)N_EXP);
    out_total[e] = nt;
    const float d  = mean - nt;
    const float sg = (d > 0.0f) ? 1.0f : ((d < 0.0f) ? -1.0f : 0.0f);
    out_bias[e] = bias[e] + BIAS_LR * sg;
}

extern "C" void kernel_launch(void* const* d_in, const int* in_sizes, int n_in,
                              void* d_out, int out_size, void* d_ws, size_t ws_size,
                              hipStream_t stream)
{
    const float* x     = (const float*)d_in[0];   // [4,4096,2048]
    const float* W     = (const float*)d_in[1];   // [2048,64]
    const float* b     = (const float*)d_in[2];   // [64]
    const float* bias  = (const float*)d_in[3];   // [64]
    const float* total = (const float*)d_in[4];   // [64]

    const int M = in_sizes[0] / N_EMBD;           // 16384 tokens

    float* out       = (float*)d_out;
    float* out_probs = out;                                   // [M,8]
    float* out_idx   = out + (size_t)M * TOPK;                // [M,8] (indices as f32)
    float* out_total = out + (size_t)2 * M * TOPK;            // [64]
    float* out_bias  = out_total + N_EXP;                     // [64]

    float* counts = (float*)d_ws;                             // [64] scratch histogram

    moe_gate_zero<<<1, N_EXP, 0, stream>>>(counts);
    moe_gate_gemm_topk<<<M / TPB, 128, 0, stream>>>(x, W, b, bias,
                                                    out_probs, out_idx, counts);
    moe_gate_finalize<<<1, N_EXP, 0, stream>>>(total, bias, counts,
                                               out_total, out_bias);
}